// Recommender_60009283059764
// MI455X (gfx1250) — compile-verified
//
#include <hip/hip_runtime.h>
#include <stdint.h>

// B=4096, L=200, V=100000, E=50
#define B_ 4096
#define L_ 200
#define E_ 50
#define XSTRIDE 256  // padded row length of x = [u(200) | emb_movie(50) | 0(6)]

// packed-weight geometry (fragment-native, K-pair interleaved)
#define K1P 252              // layer-1 K padded (250 -> 252)
#define N1P 112              // layer-1 N padded (100 -> 112)
#define W1F_ELEMS ((K1P / 2) * (2 * N1P))  // 126 * 224 = 28224
#define K2P 100
#define N2P 64               // layer-2 N padded (50 -> 64)
#define W2F_ELEMS ((K2P / 2) * (2 * N2P))  // 50 * 128 = 6400

typedef __attribute__((ext_vector_type(2))) float v2f;
typedef __attribute__((ext_vector_type(8))) float v8f;

#if __has_builtin(__builtin_amdgcn_wmma_f32_16x16x4_f32)
#define HAS_WMMA4 1
#else
#define HAS_WMMA4 0
#endif

#if HAS_WMMA4
__device__ __forceinline__ v8f wmma4(v2f a, v2f b, v8f c) {
  // D = A(16x4,f32) * B(4x16,f32) + C(16x16,f32)
  return __builtin_amdgcn_wmma_f32_16x16x4_f32(false, a, false, b, (short)0, c,
                                               false, false);
}
#else
// Stub so the host compilation pass parses; never executed on gfx1250.
__host__ __device__ static inline v8f wmma4(v2f a, v2f b, v8f c) {
  c.x += a.x * b.x + a.y * b.y;
  return c;
}
#endif

// float -> order-preserving unsigned key (ascending)
__device__ __forceinline__ unsigned int f2key(float f) {
  unsigned int u = __float_as_uint(f);
  return (u & 0x80000000u) ? ~u : (u | 0x80000000u);
}
__device__ __forceinline__ float key2f(unsigned int k) {
  unsigned int u = (k & 0x80000000u) ? (k ^ 0x80000000u) : ~k;
  return __uint_as_float(u);
}

// ---------------------------------------------------------------------------
// Kernel 0: pack W1/W2 into zero-padded, fragment-native layouts so the GEMM
// B-fragment load is one unconditional global_load_b64 per lane per K-step.
//   W1f[(k>>1)*224 + 2n + (k&1)] = (k<250 && n<100) ? W1[k*100+n] : 0
//   W2f[(k>>1)*128 + 2n + (k&1)] = (n<50)           ? W2[k*50+n]  : 0
// ---------------------------------------------------------------------------
__global__ __launch_bounds__(256) void pack_kernel(
    const float* __restrict__ W1, const float* __restrict__ W2,
    float* __restrict__ W1f, float* __restrict__ W2f) {
  const int i = blockIdx.x * 256 + threadIdx.x;
  if (i < W1F_ELEMS) {
    const int p = i / (2 * N1P);
    const int r = i - p * (2 * N1P);
    const int n = r >> 1;
    const int k = p * 2 + (r & 1);
    W1f[i] = (k < 250 && n < 100) ? W1[k * 100 + n] : 0.f;
  } else if (i < W1F_ELEMS + W2F_ELEMS) {
    const int j = i - W1F_ELEMS;
    const int p = j / (2 * N2P);
    const int r = j - p * (2 * N2P);
    const int n = r >> 1;
    const int k = p * 2 + (r & 1);
    W2f[j] = (n < 50) ? W2[k * 50 + n] : 0.f;
  }
}

// ---------------------------------------------------------------------------
// Kernel 1: one block per batch row. Gather w = emb[idx]*rating into LDS
// (feature-major, stride 201 -> conflict-free column walks), then per-feature
// masked min/max/mean and exact median via 32-pass bitwise radix select.
// Writes normalized u plus emb[movie_idx] into the padded x buffer.
// ---------------------------------------------------------------------------
__global__ __launch_bounds__(256) void stats_kernel(
    const float* __restrict__ emb, const float* __restrict__ ratings,
    const int* __restrict__ umidx, const int* __restrict__ lengths,
    const int* __restrict__ movidx, float* __restrict__ xbuf) {
  __shared__ int s_idx[L_];
  __shared__ float s_rat[L_];
  __shared__ float s_w[E_ * (L_ + 1)];  // 50 columns, stride 201 words
  __shared__ float s_red[64];
  __shared__ float s_nrm;

  const int b = blockIdx.x;
  const int t = threadIdx.x;

  int n = lengths[b];
  n = n < 1 ? 1 : (n > L_ ? L_ : n);

  for (int l = t; l < n; l += 256) {
    s_idx[l] = umidx[b * L_ + l];
    s_rat[l] = ratings[b * L_ + l];
  }
  __syncthreads();

  // gather: 50 consecutive lanes stream one embedding row
  const int total = n * E_;
  for (int i = t; i < total; i += 256) {
    int l = i / E_;
    int e = i - l * E_;
    float w = emb[s_idx[l] * E_ + e] * s_rat[l];
    s_w[e * (L_ + 1) + l] = w;
  }
  __syncthreads();

  float vmin = 0.f, vmax = 0.f, vmean = 0.f, vmed = 0.f, psq = 0.f;
  if (t < E_) {
    float* col = &s_w[t * (L_ + 1)];
    float mn = INFINITY, mx = -INFINITY, sm = 0.f;
    for (int l = 0; l < n; ++l) {
      float v = col[l];
      mn = fminf(mn, v);
      mx = fmaxf(mx, v);
      sm += v;
      col[l] = __uint_as_float(f2key(v));  // overwrite with sortable key
    }
    // k-th smallest (k = (n-1)/2) via MSB-first bit selection: O(32*n)
    const int k = (n - 1) >> 1;
    unsigned int result = 0u;
    int r = k;
    for (int bit = 31; bit >= 0; --bit) {
      const unsigned int mhi = (bit == 31) ? 0u : (0xFFFFFFFFu << (bit + 1));
      const unsigned int bm = 1u << bit;
      int c0 = 0;
      for (int l = 0; l < n; ++l) {
        unsigned int key = __float_as_uint(col[l]);
        c0 += (((key & mhi) == result) && ((key & bm) == 0u)) ? 1 : 0;
      }
      if (r >= c0) {
        result |= bm;
        r -= c0;
      }
    }
    vmin = mn;
    vmax = mx;
    vmean = sm / (float)n;
    vmed = key2f(result);
    psq = vmin * vmin + vmax * vmax + vmean * vmean + vmed * vmed;
  }
  if (t < 64) s_red[t] = (t < E_) ? psq : 0.f;
  __syncthreads();
  if (t == 0) {
    float s = 0.f;
    for (int i = 0; i < E_; ++i) s += s_red[i];
    s_nrm = rsqrtf(s);
  }
  __syncthreads();

  const float rn = s_nrm;
  float* xr = xbuf + (size_t)b * XSTRIDE;
  if (t < E_) {
    xr[t] = vmin * rn;
    xr[E_ + t] = vmax * rn;
    xr[2 * E_ + t] = vmean * rn;
    xr[3 * E_ + t] = vmed * rn;
    xr[4 * E_ + t] = emb[movidx[b] * E_ + t];  // raw (not normalized)
  }
  if (t >= 250 && t < XSTRIDE) xr[t] = 0.f;  // K padding for the GEMM
}

// ---------------------------------------------------------------------------
// Kernel 2: fused 3-layer MLP, one block per 16 batch rows, 8 waves.
// Layer1: waves 0..6 each own a 16-wide N tile of h1 (100 -> pad 112),
//         K = 250 padded to 252, f32 WMMA 16x16x4.
// Layer2: waves 0..3, N = 50 -> pad 64, K = 100.
// Layer3: 16 threads do the 50-length dot + sigmoid.
// Inner loop per K-step: ds_load_2addr (A) + global_load_b64 (B) + v_wmma.
// ---------------------------------------------------------------------------
__global__ __launch_bounds__(256) void mlp_kernel(
    const float* __restrict__ xbuf, const float* __restrict__ W1f,
    const float* __restrict__ b1, const float* __restrict__ W2f,
    const float* __restrict__ b2, const float* __restrict__ W3,
    const float* __restrict__ b3, float* __restrict__ out) {
  __shared__ float xs[16 * 257];
  __shared__ float h1[16 * 113];
  __shared__ float h2[16 * 65];

  const int t = threadIdx.x;
  const int wave = t >> 5;
  const int lane = t & 31;
  const int row0 = blockIdx.x * 16;

  for (int i = t; i < 16 * 256; i += 256) {
    int r = i >> 8, c = i & 255;
    xs[r * 257 + c] = xbuf[(size_t)(row0 + r) * XSTRIDE + c];
  }
  __syncthreads();

  // ISA 7.12.2 fragment geometry (wave32):
  //   A 16x4 : lanes 0-15 / 16-31 both M=lane%16; vgpr0/1 = K {0,1} or {2,3}
  //   B 4x16 : N = lane%16; half-wave 1 holds K+2
  //   C 16x16: N = lane%16; vgpr r holds row r (+8 for half-wave 1)
  const int hl = lane >> 4;  // half-wave
  const int koff = hl * 2;
  const int mrow0 = hl * 8;
  const int m = lane & 15;
  const int nlane = lane & 15;

  // ---- layer 1: h1 = relu(x @ W1 + b1) ----
  if (wave < 7) {
    const int ncol = wave * 16 + nlane;  // 0..111
    const float* wp = W1f + 2 * ncol;    // lane's column pair base
    v8f acc = {};
    for (int k0 = 0; k0 < K1P; k0 += 4) {
      const int ka = k0 + koff;  // even
      v2f a;
      a.x = xs[m * 257 + ka];  // cols 250..255 are zero-padded
      a.y = xs[m * 257 + ka + 1];
      const v2f bb = *(const v2f*)(wp + (ka >> 1) * (2 * N1P));
      acc = wmma4(a, bb, acc);
    }
    const float bias = (ncol < 100) ? b1[ncol] : 0.f;
    union {
      v8f v;
      float f[8];
    } u;
    u.v = acc;
    for (int r = 0; r < 8; ++r) {
      h1[(r + mrow0) * 113 + ncol] = fmaxf(u.f[r] + bias, 0.f);
    }
  }
  __syncthreads();

  // ---- layer 2: h2 = relu(h1 @ W2 + b2) ----
  if (wave < 4) {
    const int ncol = wave * 16 + nlane;  // 0..63
    const float* wp = W2f + 2 * ncol;
    v8f acc = {};
    for (int k0 = 0; k0 < K2P; k0 += 4) {
      const int ka = k0 + koff;  // <= 98, even
      v2f a;
      a.x = h1[m * 113 + ka];
      a.y = h1[m * 113 + ka + 1];
      const v2f bb = *(const v2f*)(wp + (ka >> 1) * (2 * N2P));
      acc = wmma4(a, bb, acc);
    }
    const float bias = (ncol < 50) ? b2[ncol] : 0.f;
    union {
      v8f v;
      float f[8];
    } u;
    u.v = acc;
    for (int r = 0; r < 8; ++r) {
      h2[(r + mrow0) * 65 + ncol] = fmaxf(u.f[r] + bias, 0.f);
    }
  }
  __syncthreads();

  // ---- layer 3: sigmoid(h2 @ W3 + b3) ----
  if (t < 16) {
    float z = b3[0];
    for (int j = 0; j < 50; ++j) z += h2[t * 65 + j] * W3[j];
    out[row0 + t] = 1.f / (1.f + __expf(-z));
  }
}

extern "C" void kernel_launch(void* const* d_in, const int* in_sizes, int n_in,
                              void* d_out, int out_size, void* d_ws,
                              size_t ws_size, hipStream_t stream) {
  (void)in_sizes;
  (void)n_in;
  (void)out_size;
  (void)ws_size;
  const float* emb = (const float*)d_in[0];
  const float* ratings = (const float*)d_in[1];
  const float* W1 = (const float*)d_in[2];
  const float* b1 = (const float*)d_in[3];
  const float* W2 = (const float*)d_in[4];
  const float* b2 = (const float*)d_in[5];
  const float* W3 = (const float*)d_in[6];
  const float* b3 = (const float*)d_in[7];
  const int* umidx = (const int*)d_in[8];
  const int* lengths = (const int*)d_in[9];
  const int* movidx = (const int*)d_in[10];
  float* out = (float*)d_out;

  // workspace layout: [ x : B*256 f32 | W1f : 28224 f32 | W2f : 6400 f32 ]
  float* xbuf = (float*)d_ws;                    // 4 MiB
  float* W1f = xbuf + (size_t)B_ * XSTRIDE;      // 110.25 KiB
  float* W2f = W1f + W1F_ELEMS;                  // 25 KiB

  const int pack_total = W1F_ELEMS + W2F_ELEMS;
  pack_kernel<<<(pack_total + 255) / 256, 256, 0, stream>>>(W1, W2, W1f, W2f);
  stats_kernel<<<B_, 256, 0, stream>>>(emb, ratings, umidx, lengths, movidx,
                                       xbuf);
  mlp_kernel<<<B_ / 16, 256, 0, stream>>>(xbuf, W1f, b1, W2f, b2, W3, b3, out);
}